// AttMGNN_72086731096460
// MI455X (gfx1250) — compile-verified
//
#include <hip/hip_runtime.h>
#include <hip/hip_bf16.h>
#include <math.h>

// Problem constants (from reference)
#define NN 50000      // nodes
#define NE 500000     // edges
#define NG 128        // graphs
#define DH 64         // hidden dim
#define TS 10         // message passing steps
#define WPB 8         // waves per block (256 threads, wave32)

typedef float v2f __attribute__((ext_vector_type(2)));
typedef float v8f __attribute__((ext_vector_type(8)));

// V_WMMA_F32_16X16X4_F32 : D(16x16,f32) = A(16x4,f32) x B(4x16,f32) + C
// A: lane l holds row (l&15), K = kbase + (l>>4)*2 + {0,1}  -> v2f
// B: lane l holds col (l&15), K = kbase + (l>>4)*2 + {0,1}  -> v2f
// D: lane l, reg r -> row r + 8*(l>>4), col (l&15)
__device__ inline v8f wmma4(v2f a, v2f b, v8f c) {
  return __builtin_amdgcn_wmma_f32_16x16x4_f32(
      /*neg_a=*/false, a, /*neg_b=*/false, b,
      /*c_mod=*/(short)0, c, /*reuse_a=*/false, /*reuse_b=*/false);
}

// Stage W (KTOT x 64, row-major) into LDS with K-pair interleave:
//   Wp[(k>>1)*64 + n] = (W[k][n], W[k+1][n])   (one b64 per B fragment)
template <int KTOT>
__device__ inline void stage_w(const float* __restrict__ W, float* Wlds) {
  for (int i = threadIdx.x; i < KTOT * 64; i += blockDim.x) {
    int k = i >> 6, n = i & 63;
    Wlds[(k >> 1) * 128 + n * 2 + (k & 1)] = W[i];
  }
}

__device__ inline v2f ldA(const float* p) { return *(const v2f*)p; }

// ---------------- utility kernels ----------------

__global__ void fill_kernel(float* __restrict__ p, float v, int n) {
  int i = blockIdx.x * blockDim.x + threadIdx.x;
  if (i < n) p[i] = v;
}

__global__ void counts_kernel(const int* __restrict__ batch,
                              float* __restrict__ counts) {
  int i = blockIdx.x * blockDim.x + threadIdx.x;
  if (i < NN) atomicAdd(&counts[batch[i]], 1.0f);
}

__device__ inline void atomicMaxF(float* addr, float val) {
  if (val >= 0.0f) atomicMax((int*)addr, __float_as_int(val));
  else             atomicMin((unsigned int*)addr, (unsigned int)__float_as_int(val));
}

// ---------------- h0 = x @ W_emb + b_emb  (N x 64 @ 64 x 64) ----------------

__global__ __launch_bounds__(256) void emb_kernel(
    const float* __restrict__ x, const float* __restrict__ W,
    const float* __restrict__ bias, float* __restrict__ h) {
  __shared__ __align__(16) float Wlds[64 * 64];
  stage_w<64>(W, Wlds);
  __syncthreads();
  const v2f* Wp = (const v2f*)Wlds;

  const int tile = blockIdx.x * WPB + (threadIdx.x >> 5);
  if (tile >= NN / 16) return;
  const int lane = threadIdx.x & 31;
  const int half = lane >> 4, lo = lane & 15;
  const int nbase = tile * 16;
  const float* xr = x + (size_t)(nbase + lo) * 64;

  v8f acc[4] = {};
  for (int kb = 0; kb < 64; kb += 4) {
    const int ka = kb + half * 2;
    v2f a = ldA(xr + ka);
#pragma unroll
    for (int nt = 0; nt < 4; ++nt)
      acc[nt] = wmma4(a, Wp[(ka >> 1) * 64 + nt * 16 + lo], acc[nt]);
  }
#pragma unroll
  for (int nt = 0; nt < 4; ++nt) {
    const int col = nt * 16 + lo;
    const float bs = bias[col];
#pragma unroll
    for (int r = 0; r < 8; ++r)
      h[(size_t)(nbase + r + 8 * half) * 64 + col] = acc[nt][r] + bs;
  }
}

// ---- m = relu(concat(h[src],h[dst],ea) @ W_msg + b), scatter-add to agg[dst]

__global__ __launch_bounds__(256) void msg_kernel(
    const float* __restrict__ h, const int* __restrict__ ei,
    const float* __restrict__ ea, const float* __restrict__ W,
    const float* __restrict__ bias, float* __restrict__ agg) {
  __shared__ __align__(16) float Wlds[136 * 64];
  stage_w<136>(W, Wlds);
  __syncthreads();
  const v2f* Wp = (const v2f*)Wlds;

  const int tile = blockIdx.x * WPB + (threadIdx.x >> 5);
  if (tile >= NE / 16) return;
  const int lane = threadIdx.x & 31;
  const int half = lane >> 4, lo = lane & 15;
  const int ebase = tile * 16;
  const int erow = ebase + lo;
  const int s = ei[erow];
  const int d = ei[NE + erow];
  const float* hs = h + (size_t)s * 64;
  const float* hd = h + (size_t)d * 64;
  const float* er = ea + (size_t)erow * 8;

  v8f acc[4] = {};
  for (int kb = 0; kb < 136; kb += 4) {
    const int ka = kb + half * 2;
    v2f a;
    if (ka < 64)       a = ldA(hs + ka);
    else if (ka < 128) a = ldA(hd + ka - 64);
    else               a = ldA(er + ka - 128);
#pragma unroll
    for (int nt = 0; nt < 4; ++nt)
      acc[nt] = wmma4(a, Wp[(ka >> 1) * 64 + nt * 16 + lo], acc[nt]);
  }

  int dstm[8];
#pragma unroll
  for (int r = 0; r < 8; ++r) dstm[r] = ei[NE + ebase + r + 8 * half];

#pragma unroll
  for (int nt = 0; nt < 4; ++nt) {
    const int col = nt * 16 + lo;
    const float bs = bias[col];
#pragma unroll
    for (int r = 0; r < 8; ++r) {
      float m = acc[nt][r] + bs;
      m = m > 0.0f ? m : 0.0f;
      atomicAdd(&agg[(size_t)dstm[r] * 64 + col], m);
    }
  }
}

// ---- h_new = relu(concat(h,agg) @ W_upd + b); h <- h_new; pooled[batch] += h_new

__global__ __launch_bounds__(256) void upd_kernel(
    float* __restrict__ h, const float* __restrict__ agg,
    const int* __restrict__ batch, const float* __restrict__ W,
    const float* __restrict__ bias, float* __restrict__ pooled) {
  __shared__ __align__(16) float Wlds[128 * 64];
  stage_w<128>(W, Wlds);
  __syncthreads();
  const v2f* Wp = (const v2f*)Wlds;

  const int tile = blockIdx.x * WPB + (threadIdx.x >> 5);
  if (tile >= NN / 16) return;
  const int lane = threadIdx.x & 31;
  const int half = lane >> 4, lo = lane & 15;
  const int nbase = tile * 16;
  const float* hr = h + (size_t)(nbase + lo) * 64;
  const float* ar = agg + (size_t)(nbase + lo) * 64;

  v8f acc[4] = {};
  for (int kb = 0; kb < 128; kb += 4) {
    const int ka = kb + half * 2;
    v2f a = (ka < 64) ? ldA(hr + ka) : ldA(ar + ka - 64);
#pragma unroll
    for (int nt = 0; nt < 4; ++nt)
      acc[nt] = wmma4(a, Wp[(ka >> 1) * 64 + nt * 16 + lo], acc[nt]);
  }

  int bm[8];
#pragma unroll
  for (int r = 0; r < 8; ++r) bm[r] = batch[nbase + r + 8 * half];

#pragma unroll
  for (int nt = 0; nt < 4; ++nt) {
    const int col = nt * 16 + lo;
    const float bs = bias[col];
#pragma unroll
    for (int r = 0; r < 8; ++r) {
      float hn = acc[nt][r] + bs;
      hn = hn > 0.0f ? hn : 0.0f;
      h[(size_t)(nbase + r + 8 * half) * 64 + col] = hn;
      atomicAdd(&pooled[(size_t)bm[r] * 64 + col], hn);
    }
  }
}

// ---- conf_t = sigmoid((pooled . W_conf)/count + b); ew = exp(conf); den += ew

__global__ void conf_kernel(const float* __restrict__ pooled,
                            const float* __restrict__ counts,
                            const float* __restrict__ Wc,
                            const float* __restrict__ bc,
                            float* __restrict__ ew, float* __restrict__ hden) {
  int g = threadIdx.x;
  if (g < NG) {
    float cnt = fmaxf(counts[g], 1.0f);
    float s = 0.0f;
    for (int c = 0; c < 64; ++c) s += pooled[g * 64 + c] * Wc[c];
    s = s / cnt + bc[0];
    float conf = 1.0f / (1.0f + expf(-s));
    float e = expf(conf);           // softmax-over-T numerator (conf in (0,1): safe)
    ew[g] = e;
    hden[g] += e;
  }
}

// ---- hid_num += ew[batch] * h_new (online softmax-weighted sum over T)

__global__ void accum_kernel(const float* __restrict__ h,
                             const int* __restrict__ batch,
                             const float* __restrict__ ew,
                             float* __restrict__ hid_num) {
  int i = blockIdx.x * blockDim.x + threadIdx.x;
  if (i < NN * 64) hid_num[i] += ew[batch[i >> 6]] * h[i];
}

__global__ void hidden_kernel(const float* __restrict__ hid_num,
                              const float* __restrict__ hden,
                              const int* __restrict__ batch,
                              float* __restrict__ hidden) {
  int i = blockIdx.x * blockDim.x + threadIdx.x;
  if (i < NN * 64) hidden[i] = hid_num[i] / hden[batch[i >> 6]];
}

// ---- score = tanh(concat(x,hidden) @ W_att1 + b1) @ w_att2 + b2

__global__ __launch_bounds__(256) void att_kernel(
    const float* __restrict__ x, const float* __restrict__ hidden,
    const float* __restrict__ W, const float* __restrict__ b1,
    const float* __restrict__ w2, const float* __restrict__ b2,
    float* __restrict__ score) {
  __shared__ __align__(16) float Wlds[128 * 64];
  stage_w<128>(W, Wlds);
  __syncthreads();
  const v2f* Wp = (const v2f*)Wlds;

  const int tile = blockIdx.x * WPB + (threadIdx.x >> 5);
  if (tile >= NN / 16) return;
  const int lane = threadIdx.x & 31;
  const int half = lane >> 4, lo = lane & 15;
  const int nbase = tile * 16;
  const float* xr = x + (size_t)(nbase + lo) * 64;
  const float* hr = hidden + (size_t)(nbase + lo) * 64;

  v8f acc[4] = {};
  for (int kb = 0; kb < 128; kb += 4) {
    const int ka = kb + half * 2;
    v2f a = (ka < 64) ? ldA(xr + ka) : ldA(hr + ka - 64);
#pragma unroll
    for (int nt = 0; nt < 4; ++nt)
      acc[nt] = wmma4(a, Wp[(ka >> 1) * 64 + nt * 16 + lo], acc[nt]);
  }

  float part[8];
#pragma unroll
  for (int r = 0; r < 8; ++r) part[r] = 0.0f;
#pragma unroll
  for (int nt = 0; nt < 4; ++nt) {
    const int col = nt * 16 + lo;
    const float ba = b1[col];
    const float wa = w2[col];
#pragma unroll
    for (int r = 0; r < 8; ++r)
      part[r] += tanhf(acc[nt][r] + ba) * wa;
  }
  // reduce across the 16 lanes of each half-wave (cols 0..15)
#pragma unroll
  for (int m = 1; m < 16; m <<= 1) {
#pragma unroll
    for (int r = 0; r < 8; ++r) part[r] += __shfl_xor(part[r], m, 32);
  }
  if (lo == 0) {
    const float bb = b2[0];
#pragma unroll
    for (int r = 0; r < 8; ++r)
      score[nbase + r + 8 * half] = part[r] + bb;
  }
}

// ---- segment softmax + attention pooling ----

__global__ void smax_kernel(const float* __restrict__ score,
                            const int* __restrict__ batch,
                            float* __restrict__ smax) {
  int i = blockIdx.x * blockDim.x + threadIdx.x;
  if (i < NN) atomicMaxF(&smax[batch[i]], score[i]);
}

__global__ void es_kernel(const float* __restrict__ score,
                          const int* __restrict__ batch,
                          const float* __restrict__ smax,
                          float* __restrict__ es, float* __restrict__ denom) {
  int i = blockIdx.x * blockDim.x + threadIdx.x;
  if (i < NN) {
    float e = expf(score[i] - smax[batch[i]]);
    es[i] = e;
    atomicAdd(&denom[batch[i]], e);
  }
}

__global__ void gfeat_kernel(const float* __restrict__ es,
                             const float* __restrict__ denom,
                             const float* __restrict__ hidden,
                             const int* __restrict__ batch,
                             float* __restrict__ gfeat) {
  int i = blockIdx.x * blockDim.x + threadIdx.x;
  if (i < NN * 64) {
    int n = i >> 6, c = i & 63;
    int g = batch[n];
    float alpha = es[n] / fmaxf(denom[g], 1e-16f);
    atomicAdd(&gfeat[g * 64 + c], alpha * hidden[i]);
  }
}

__global__ void head_kernel(const float* __restrict__ gfeat,
                            const float* __restrict__ counts,
                            const float* __restrict__ Wh,
                            const float* __restrict__ bh,
                            float* __restrict__ out) {
  int g = threadIdx.x;
  if (g < NG) {
    float s = 0.0f;
    for (int c = 0; c < 64; ++c) s += gfeat[g * 64 + c] * Wh[c];
    out[g] = s + bh[0];
    out[NG + g] = counts[g];   // second tuple output
  }
}

// ---------------- launcher ----------------

extern "C" void kernel_launch(void* const* d_in, const int* in_sizes, int n_in,
                              void* d_out, int out_size, void* d_ws, size_t ws_size,
                              hipStream_t stream) {
  const float* x      = (const float*)d_in[0];
  const int*   ei     = (const int*)  d_in[1];
  const float* ea     = (const float*)d_in[2];
  const int*   batch  = (const int*)  d_in[3];
  const float* W_emb  = (const float*)d_in[4];
  const float* b_emb  = (const float*)d_in[5];
  const float* W_msg  = (const float*)d_in[6];
  const float* b_msg  = (const float*)d_in[7];
  const float* W_upd  = (const float*)d_in[8];
  const float* b_upd  = (const float*)d_in[9];
  const float* W_conf = (const float*)d_in[10];
  const float* b_conf = (const float*)d_in[11];
  const float* W_att1 = (const float*)d_in[12];
  const float* b_att1 = (const float*)d_in[13];
  const float* w_att2 = (const float*)d_in[14];
  const float* b_att2 = (const float*)d_in[15];
  const float* W_head = (const float*)d_in[16];
  const float* b_head = (const float*)d_in[17];
  float* out = (float*)d_out;

  float* ws = (float*)d_ws;
  float* h       = ws; ws += (size_t)NN * DH;   // 12.8 MB
  float* agg     = ws; ws += (size_t)NN * DH;
  float* hid_num = ws; ws += (size_t)NN * DH;
  float* hidden  = ws; ws += (size_t)NN * DH;
  float* score   = ws; ws += NN;
  float* esv     = ws; ws += NN;
  float* counts  = ws; ws += NG;
  float* hid_den = ws; ws += NG;
  float* pooled  = ws; ws += (size_t)NG * DH;
  float* ewb     = ws; ws += NG;
  float* smax    = ws; ws += NG;
  float* denom   = ws; ws += NG;
  float* gfeat   = ws; ws += (size_t)NG * DH;

  auto fill = [&](float* p, float v, int n) {
    fill_kernel<<<(n + 255) / 256, 256, 0, stream>>>(p, v, n);
  };

  const int nodeTiles = NN / 16;                 // 3125 (exact)
  const int edgeTiles = NE / 16;                 // 31250 (exact)
  const int nodeBlocks = (nodeTiles + WPB - 1) / WPB;
  const int edgeBlocks = (edgeTiles + WPB - 1) / WPB;

  fill(counts, 0.0f, NG);
  fill(hid_num, 0.0f, NN * DH);
  fill(hid_den, 0.0f, NG);
  counts_kernel<<<(NN + 255) / 256, 256, 0, stream>>>(batch, counts);
  emb_kernel<<<nodeBlocks, 32 * WPB, 0, stream>>>(x, W_emb, b_emb, h);

  for (int t = 0; t < TS; ++t) {
    fill(agg, 0.0f, NN * DH);
    fill(pooled, 0.0f, NG * DH);
    msg_kernel<<<edgeBlocks, 32 * WPB, 0, stream>>>(h, ei, ea, W_msg, b_msg, agg);
    upd_kernel<<<nodeBlocks, 32 * WPB, 0, stream>>>(h, agg, batch, W_upd, b_upd, pooled);
    conf_kernel<<<1, 128, 0, stream>>>(pooled, counts, W_conf, b_conf, ewb, hid_den);
    accum_kernel<<<(NN * DH + 255) / 256, 256, 0, stream>>>(h, batch, ewb, hid_num);
  }

  hidden_kernel<<<(NN * DH + 255) / 256, 256, 0, stream>>>(hid_num, hid_den, batch, hidden);
  fill(smax, -3.402823466e38f, NG);
  fill(denom, 0.0f, NG);
  fill(gfeat, 0.0f, NG * DH);
  att_kernel<<<nodeBlocks, 32 * WPB, 0, stream>>>(x, hidden, W_att1, b_att1, w_att2, b_att2, score);
  smax_kernel<<<(NN + 255) / 256, 256, 0, stream>>>(score, batch, smax);
  es_kernel<<<(NN + 255) / 256, 256, 0, stream>>>(score, batch, smax, esv, denom);
  gfeat_kernel<<<(NN * DH + 255) / 256, 256, 0, stream>>>(esv, denom, hidden, batch, gfeat);
  head_kernel<<<1, 128, 0, stream>>>(gfeat, counts, W_head, b_head, out);
}